// LocalSparseAttention_27779848471221
// MI455X (gfx1250) — compile-verified
//
#include <hip/hip_runtime.h>
#include <hip/hip_bf16.h>
#include <stdint.h>

typedef __attribute__((ext_vector_type(2))) float v2f;
typedef __attribute__((ext_vector_type(8))) float v8f;

#define WMMA_F32(a, b, c) \
    __builtin_amdgcn_wmma_f32_16x16x4_f32(false, (a), false, (b), (short)0, (c), false, false)

// ---------------------------------------------------------------------------
// Problem constants (from reference)
// ---------------------------------------------------------------------------
#define BB 2
#define SS 2048
#define DD 1024
#define HH 16
#define HD 64
#define HALF_WIN 64          // WINDOW // 2
#define N3 (3 * DD)          // 3072
#define NTOK (BB * SS)       // 4096
#define SCALE 0.125f         // HD^-0.5

// Async copy global -> LDS, 16 bytes per lane. ldsOff = low 32 bits of the
// generic LDS pointer (flat->LDS truncation: addr[31:0] is the LDS offset).
__device__ __forceinline__ void async_copy_b128(uint32_t ldsOff, const float* gptr) {
    uint64_t gaddr = (uint64_t)(uintptr_t)gptr;
    asm volatile("global_load_async_to_lds_b128 %0, %1, off"
                 :: "v"(ldsOff), "v"(gaddr) : "memory");
}
__device__ __forceinline__ uint32_t lds_off(const void* p) {
    return (uint32_t)(uintptr_t)p;
}

// ---------------------------------------------------------------------------
// C[M,N] = A[M,K] @ W[K,N] + bias[N], fp32 WMMA 16x16x4.
// Block: 256 threads (8 waves). Block tile 128(M) x 256(N); each wave owns a
// 64x64 sub-tile (16 C-tiles). K chunked by 16, double-buffered LDS filled by
// GLOBAL_LOAD_ASYNC_TO_LDS_B128 issued one chunk ahead (ASYNCcnt pipelining).
// ---------------------------------------------------------------------------
#define LDA 20   // padded A-chunk row (floats): 80B rows -> 16B aligned, conflict-free frags

__global__ __launch_bounds__(256) void gemm_bias_kernel(
    const float* __restrict__ A, const float* __restrict__ W,
    const float* __restrict__ bias, float* __restrict__ C,
    int M, int N, int K)
{
    __shared__ float lA[2][128 * LDA];   // [buf][row 0..127][k 0..15], padded
    __shared__ float lB[2][16 * 256];    // [buf][k 0..15][n 0..255]

    const int tid   = threadIdx.x;
    const int lane  = tid & 31;
    const int wave  = tid >> 5;
    const int half  = lane >> 4;       // 0: lanes 0-15, 1: lanes 16-31
    const int l16   = lane & 15;
    const int waveM = (wave >> 2) * 64;  // 0 or 64
    const int waveN = (wave & 3) * 64;   // 0,64,128,192

    const int mBase = blockIdx.y * 128;
    const int nBase = blockIdx.x * 256;

    v8f acc[4][4];
    for (int mt = 0; mt < 4; ++mt)
        for (int nt = 0; nt < 4; ++nt)
            acc[mt][nt] = (v8f){0, 0, 0, 0, 0, 0, 0, 0};

    // Issue async loads for one 16-wide K chunk into buffer `buf`:
    // A: 128x16 floats = 512 B128s (2/thread); W: 16x256 floats = 1024 B128s (4/thread).
    auto issue_chunk = [&](int k0, int buf) {
#pragma unroll
        for (int i = 0; i < 2; ++i) {
            int f   = i * 256 + tid;
            int row = f >> 2;
            int seg = (f & 3) * 4;
            async_copy_b128(lds_off(&lA[buf][row * LDA + seg]),
                            A + (size_t)(mBase + row) * K + k0 + seg);
        }
#pragma unroll
        for (int i = 0; i < 4; ++i) {
            int f    = i * 256 + tid;
            int krow = f >> 6;
            int col  = (f & 63) * 4;
            async_copy_b128(lds_off(&lB[buf][krow * 256 + col]),
                            W + (size_t)(k0 + krow) * N + nBase + col);
        }
    };

    issue_chunk(0, 0);
    int p = 0;

    for (int k0 = 0; k0 < K; k0 += 16) {
        __syncthreads();   // all waves done reading buf 1-p (previous compute)
        if (k0 + 16 < K) {
            issue_chunk(k0 + 16, 1 - p);
            // 6 new in flight; in-order completion => prior chunk's 6 are done
            asm volatile("s_wait_asynccnt 0x6" ::: "memory");
        } else {
            asm volatile("s_wait_asynccnt 0x0" ::: "memory");
        }
        __syncthreads();   // publish buf p across waves

        const float* bufA = lA[p];
        const float* bufB = lB[p];
#pragma unroll
        for (int kk = 0; kk < 4; ++kk) {
            const int kc = kk * 4 + half * 2;   // this lane's K pair
            v2f afrag[4];
#pragma unroll
            for (int mt = 0; mt < 4; ++mt)
                afrag[mt] = *(const v2f*)&bufA[(waveM + mt * 16 + l16) * LDA + kc];
            v2f bfrag[4];
#pragma unroll
            for (int nt = 0; nt < 4; ++nt) {
                int n = waveN + nt * 16 + l16;
                v2f b;
                b[0] = bufB[(kc + 0) * 256 + n];  // vgpr0: rows K, K+2 by half
                b[1] = bufB[(kc + 1) * 256 + n];  // vgpr1: rows K+1, K+3
                bfrag[nt] = b;
            }
#pragma unroll
            for (int mt = 0; mt < 4; ++mt)
#pragma unroll
                for (int nt = 0; nt < 4; ++nt)
                    acc[mt][nt] = WMMA_F32(afrag[mt], bfrag[nt], acc[mt][nt]);
        }
        p ^= 1;
    }

    // Store with bias. C layout: vgpr v -> row v + 8*half, col l16.
    for (int nt = 0; nt < 4; ++nt) {
        int n = nBase + waveN + nt * 16 + l16;
        float bv = bias[n];
        for (int mt = 0; mt < 4; ++mt) {
            for (int v = 0; v < 8; ++v) {
                int m = mBase + waveM + mt * 16 + v + 8 * half;
                C[(size_t)m * N + n] = acc[mt][nt][v] + bv;
            }
        }
    }
}

// ---------------------------------------------------------------------------
// Banded attention. One wave per 16-query tile; 2 waves per block.
// qkv: [NTOK][3*D] with columns (section, head, hd): q @ h*64, k @ D+h*64,
// v @ 2D+h*64. Output attn: [NTOK][D], column h*64+d.
// ---------------------------------------------------------------------------
#define LDT 68     // padded row length for 16x64 Q/K/V tiles (272B rows)
#define NKT 10     // key tiles covering [q0-64, q0+96)
#define LDP 164    // padded row length for 16x160 score buffer (656B rows)

__global__ __launch_bounds__(64) void attention_kernel(
    const float* __restrict__ qkv, float* __restrict__ attn)
{
    __shared__ float sQ[2][16 * LDT];
    __shared__ float sKV[2][16 * LDT];
    __shared__ float sP[2][16 * LDP];

    const int lane = threadIdx.x & 31;
    const int wv   = threadIdx.x >> 5;
    const int half = lane >> 4;
    const int l16  = lane & 15;

    const int w   = blockIdx.x * 2 + wv;       // global query-tile id
    const int b   = w >> 11;                   // / (H * S/16) = / 2048
    const int rem = w & 2047;
    const int h   = rem >> 7;                  // / 128
    const int q0  = (rem & 127) * 16;

    const size_t tokBase = (size_t)b * SS;
    float* Q  = sQ[wv];
    float* KV = sKV[wv];
    float* P  = sP[wv];

    // ---- Load Q tile (16 x 64) into LDS: 256 float4, 8 per lane ----
    {
        const size_t colQ = (size_t)h * HD;
        for (int i = 0; i < 8; ++i) {
            int f   = i * 32 + lane;
            int row = f >> 4;
            int seg = (f & 15) * 4;
            float4 qv = *(const float4*)(qkv + (tokBase + q0 + row) * N3 + colQ + seg);
            *(float4*)&Q[row * LDT + seg] = qv;
        }
    }

    float rowmax[8];
    for (int v = 0; v < 8; ++v) rowmax[v] = -__builtin_inff();

    // ---- Pass 1: scores S = scale * Q K^T + band mask, parked in LDS P ----
    for (int kt = 0; kt < NKT; ++kt) {
        const int k0 = q0 - HALF_WIN + kt * 16;
        const bool validTile = (k0 >= 0) && (k0 < SS);
        if (!validTile) {
            for (int v = 0; v < 8; ++v)
                P[(v + 8 * half) * LDP + kt * 16 + l16] = -__builtin_inff();
            continue;
        }
        const size_t colK = (size_t)DD + (size_t)h * HD;
        for (int i = 0; i < 8; ++i) {
            int f   = i * 32 + lane;
            int row = f >> 4;
            int seg = (f & 15) * 4;
            float4 kvv = *(const float4*)(qkv + (tokBase + k0 + row) * N3 + colK + seg);
            *(float4*)&KV[row * LDT + seg] = kvv;
        }

        v8f acc = (v8f){0, 0, 0, 0, 0, 0, 0, 0};
        for (int kk = 0; kk < 16; ++kk) {        // 64 dims / 4
            const int kc = kk * 4 + half * 2;
            v2f a  = *(const v2f*)&Q[l16 * LDT + kc];   // A: lane=M(query)
            v2f bq = *(const v2f*)&KV[l16 * LDT + kc];  // B=K^T: lane=N(token)
            acc = WMMA_F32(a, bq, acc);
        }
        const int k = k0 + l16;
        for (int v = 0; v < 8; ++v) {
            const int q = q0 + v + 8 * half;
            float s = acc[v] * SCALE;
            int d = q - k;
            if (d < 0) d = -d;
            if (d > HALF_WIN) s = -__builtin_inff();
            rowmax[v] = fmaxf(rowmax[v], s);
            P[(v + 8 * half) * LDP + kt * 16 + l16] = s;
        }
    }

    // ---- Row max across the 16 lanes of each half ----
    for (int v = 0; v < 8; ++v)
        for (int off = 8; off >= 1; off >>= 1)
            rowmax[v] = fmaxf(rowmax[v], __shfl_xor(rowmax[v], off, 16));

    // ---- Pass 2: exponentiate in place, accumulate row sums ----
    float rowsum[8];
    for (int v = 0; v < 8; ++v) rowsum[v] = 0.0f;
    for (int kt = 0; kt < NKT; ++kt) {
        for (int v = 0; v < 8; ++v) {
            const int idx = (v + 8 * half) * LDP + kt * 16 + l16;
            float pexp = __expf(P[idx] - rowmax[v]);   // -inf -> 0
            P[idx] = pexp;
            rowsum[v] += pexp;
        }
    }
    for (int v = 0; v < 8; ++v)
        for (int off = 8; off >= 1; off >>= 1)
            rowsum[v] += __shfl_xor(rowsum[v], off, 16);
    float rinv[8];
    for (int v = 0; v < 8; ++v) rinv[v] = 1.0f / rowsum[v];

    // ---- O = P @ V (16 x 64), 4 N-tiles ----
    v8f oacc[4];
    for (int nt = 0; nt < 4; ++nt) oacc[nt] = (v8f){0, 0, 0, 0, 0, 0, 0, 0};

    for (int kt = 0; kt < NKT; ++kt) {
        const int k0 = q0 - HALF_WIN + kt * 16;
        if (k0 < 0 || k0 >= SS) continue;
        const size_t colV = (size_t)(2 * DD) + (size_t)h * HD;
        for (int i = 0; i < 8; ++i) {
            int f   = i * 32 + lane;
            int row = f >> 4;
            int seg = (f & 15) * 4;
            float4 vvv = *(const float4*)(qkv + (tokBase + k0 + row) * N3 + colV + seg);
            *(float4*)&KV[row * LDT + seg] = vvv;
        }
        for (int kk = 0; kk < 4; ++kk) {          // 16 tokens / 4
            const int tokc = kk * 4 + half * 2;   // this lane's token pair
            v2f a = *(const v2f*)&P[l16 * LDP + kt * 16 + tokc]; // A=P: lane=M(query)
            for (int nt = 0; nt < 4; ++nt) {
                int n = nt * 16 + l16;
                v2f bvv;
                bvv[0] = KV[(tokc + 0) * LDT + n];  // B=V: rows=token, cols=dim
                bvv[1] = KV[(tokc + 1) * LDT + n];
                oacc[nt] = WMMA_F32(a, bvv, oacc[nt]);
            }
        }
    }

    // ---- Store O (normalized) into attn[token][h*64 + d] ----
    for (int nt = 0; nt < 4; ++nt) {
        const int col = h * HD + nt * 16 + l16;
        for (int v = 0; v < 8; ++v) {
            const int q = q0 + v + 8 * half;
            attn[(tokBase + q) * DD + col] = oacc[nt][v] * rinv[v];
        }
    }
}

// ---------------------------------------------------------------------------
// Host launcher
// ---------------------------------------------------------------------------
extern "C" void kernel_launch(void* const* d_in, const int* in_sizes, int n_in,
                              void* d_out, int out_size, void* d_ws, size_t ws_size,
                              hipStream_t stream) {
    const float* x     = (const float*)d_in[0];
    const float* w_qkv = (const float*)d_in[1];
    const float* b_qkv = (const float*)d_in[2];
    const float* w_out = (const float*)d_in[3];
    const float* b_out = (const float*)d_in[4];
    float* out = (float*)d_out;

    float* qkv  = (float*)d_ws;                       // [4096][3072]
    float* attn = qkv + (size_t)NTOK * N3;            // [4096][1024]

    // QKV projection: [4096,1024] x [1024,3072] + bias
    gemm_bias_kernel<<<dim3(N3 / 256, NTOK / 128), dim3(256), 0, stream>>>(
        x, w_qkv, b_qkv, qkv, NTOK, N3, DD);

    // Banded attention: 4096 query tiles, 2 waves/block
    attention_kernel<<<dim3((BB * HH * (SS / 16)) / 2), dim3(64), 0, stream>>>(
        qkv, attn);

    // Output projection: [4096,1024] x [1024,1024] + bias
    gemm_bias_kernel<<<dim3(DD / 256, NTOK / 128), dim3(256), 0, stream>>>(
        attn, w_out, b_out, out, NTOK, DD, DD);
}